// TransLayer_57492432224641
// MI455X (gfx1250) — compile-verified
//
#include <hip/hip_runtime.h>

typedef float v2f __attribute__((ext_vector_type(2)));
typedef float v8f __attribute__((ext_vector_type(8)));

static constexpr int kH    = 8;
static constexpr int kDH   = 64;
static constexpr int kDIM  = 512;
static constexpr int kM    = 256;    // landmarks
static constexpr int kN    = 8192;   // tokens
static constexpr int kB    = 2;
static constexpr int kBH   = 16;     // b*h
static constexpr int kKER  = 33;

static __device__ __forceinline__ v8f wmma_f32(v2f a, v2f b, v8f c) {
  // D = A(16x4) * B(4x16) + C, fp32 matrix core
  return __builtin_amdgcn_wmma_f32_16x16x4_f32(false, a, false, b, (short)0, c, false, false);
}

// CDNA5 async global->LDS copies (ASYNCcnt path, no VGPR bounce).
// Flat pointers to __shared__ carry the LDS byte offset in their low 32 bits
// (shared aperture base occupies addr[63:32] only).
static __device__ __forceinline__ void async_load_b128(const void* gaddr, void* lds) {
  unsigned l = (unsigned)(unsigned long long)lds;
  asm volatile("global_load_async_to_lds_b128 %0, %1, off"
               :: "v"(l), "v"(gaddr) : "memory");
}
static __device__ __forceinline__ void async_load_b32(const void* gaddr, void* lds) {
  unsigned l = (unsigned)(unsigned long long)lds;
  asm volatile("global_load_async_to_lds_b32 %0, %1, off"
               :: "v"(l), "v"(gaddr) : "memory");
}
static __device__ __forceinline__ void wait_async0() {
  asm volatile("s_wait_asynccnt 0x0" ::: "memory");
}

// ---------------- LayerNorm: one wave per token row (512 elems) ----------------
__global__ __launch_bounds__(256) void ln_kernel(
    const float* __restrict__ x, const float* __restrict__ g,
    const float* __restrict__ bta, float* __restrict__ xn) {
  int wave = threadIdx.x >> 5, lane = threadIdx.x & 31;
  int tok = blockIdx.x * 8 + wave;
  const float* row = x + (size_t)tok * kDIM;
  float v[16], s = 0.f, ss = 0.f;
  for (int i = 0; i < 16; ++i) { v[i] = row[lane + 32 * i]; s += v[i]; ss += v[i] * v[i]; }
  for (int off = 16; off >= 1; off >>= 1) {
    s += __shfl_xor(s, off, 32);
    ss += __shfl_xor(ss, off, 32);
  }
  float mu = s * (1.0f / kDIM);
  float var = ss * (1.0f / kDIM) - mu * mu;
  float inv = rsqrtf(var + 1e-5f);
  float* o = xn + (size_t)tok * kDIM;
  for (int i = 0; i < 16; ++i) {
    int c = lane + 32 * i;
    o[c] = (v[i] - mu) * inv * g[c] + bta[c];
  }
}

// ---------------- Generic batched WMMA-f32 GEMM ----------------
// C[M,N] = alpha * A[M,K] @ B[K,N]  (+bias[col]) (+res[row,col])
// M,N multiples of 64; K multiple of 16. Batch via blockIdx.z with strides.
// Tiles staged via async global->LDS (b128, 16B-aligned everywhere).
__global__ __launch_bounds__(256) void gemm_wmma(
    const float* __restrict__ A, const float* __restrict__ B,
    float* __restrict__ C, const float* __restrict__ bias,
    const float* __restrict__ res,
    int M, int N, int K, int lda, int ldb, int ldc, int ldr,
    long sA, long sB, long sC, float alpha) {
  __shared__ float As[64][20];   // row stride 80B (16B-aligned)
  __shared__ float Bs[16][68];   // row stride 272B (16B-aligned)
  int z = blockIdx.z;
  A += (long)z * sA; B += (long)z * sB; C += (long)z * sC;
  int bm = blockIdx.y * 64, bn = blockIdx.x * 64;
  int tid = threadIdx.x, lane = tid & 31;
  int w = tid >> 5, wm = w & 3, wn = w >> 2;
  int l15 = lane & 15, hi = (lane >> 4) & 1;
  int ra = tid >> 2, ca = (tid & 3) * 4;    // A-tile 64x16: one b128 per thread
  int rb = tid >> 4, cb = (tid & 15) * 4;   // B-tile 16x64: one b128 per thread
  v8f acc0 = {}, acc1 = {};
  for (int k0 = 0; k0 < K; k0 += 16) {
    {
      const float* ap = A + (size_t)(bm + ra) * lda + k0 + ca;
      async_load_b128(ap, &As[ra][ca]);
      if (k0 + 16 < K) __builtin_prefetch(ap + 16, 0, 1);  // global_prefetch_b8
    }
    {
      const float* bp = B + (size_t)(k0 + rb) * ldb + bn + cb;
      async_load_b128(bp, &Bs[rb][cb]);
    }
    wait_async0();
    __syncthreads();
    int arow = wm * 16 + l15;
    int bc0 = wn * 32 + l15;
    for (int kk = 0; kk < 4; ++kk) {
      int ks = kk * 4 + hi * 2;
      v2f af = { As[arow][ks], As[arow][ks + 1] };
      v2f b0 = { Bs[ks][bc0], Bs[ks + 1][bc0] };
      acc0 = wmma_f32(af, b0, acc0);
      v2f b1 = { Bs[ks][bc0 + 16], Bs[ks + 1][bc0 + 16] };
      acc1 = wmma_f32(af, b1, acc1);
    }
    __syncthreads();
  }
  for (int r = 0; r < 8; ++r) {
    int grow = bm + wm * 16 + r + hi * 8;
    int gc0 = bn + wn * 32 + l15;
    float v0 = acc0[r] * alpha, v1 = acc1[r] * alpha;
    if (bias) { v0 += bias[gc0]; v1 += bias[gc0 + 16]; }
    if (res)  { v0 += res[(size_t)grow * ldr + gc0]; v1 += res[(size_t)grow * ldr + gc0 + 16]; }
    C[(size_t)grow * ldc + gc0] = v0;
    C[(size_t)grow * ldc + gc0 + 16] = v1;
  }
}

// ---------------- landmarks: q_l (scaled), k_l, k_l^T ----------------
__global__ __launch_bounds__(256) void landmark_kernel(
    const float* __restrict__ qkv, float* __restrict__ ql,
    float* __restrict__ kl, float* __restrict__ klt) {
  int idx = blockIdx.x * blockDim.x + threadIdx.x;
  int d = idx & 63, m = (idx >> 6) & 255, z = idx >> 14;
  int b = z >> 3, h = z & 7;
  const float* qp = qkv + ((size_t)(b * kN + m * 32)) * (3 * kDIM) + h * kDH + d;
  const float* kp = qp + kDIM;
  float sq = 0.f, sk = 0.f;
  for (int j = 0; j < 32; ++j) {
    sq += qp[(size_t)j * 3 * kDIM];
    sk += kp[(size_t)j * 3 * kDIM];
  }
  float qv = sq * (1.0f / 32.0f) * 0.125f;   // includes DH^-0.5
  float kv = sk * (1.0f / 32.0f);
  ql[(size_t)z * kM * kDH + m * kDH + d] = qv;
  kl[(size_t)z * kM * kDH + m * kDH + d] = kv;
  klt[(size_t)z * kM * kDH + d * kM + m] = kv;
}

// ---------------- softmax over rows of width 256 (one wave/row) ----------------
__global__ __launch_bounds__(256) void softmax256_kernel(float* __restrict__ a) {
  int wave = threadIdx.x >> 5, lane = threadIdx.x & 31;
  int row = blockIdx.x * 8 + wave;
  float* p = a + (size_t)row * 256;
  float v[8], mx = -1e30f;
  for (int i = 0; i < 8; ++i) { v[i] = p[lane + 32 * i]; mx = fmaxf(mx, v[i]); }
  for (int off = 16; off >= 1; off >>= 1) mx = fmaxf(mx, __shfl_xor(mx, off, 32));
  float s = 0.f;
  for (int i = 0; i < 8; ++i) { v[i] = __expf(v[i] - mx); s += v[i]; }
  for (int off = 16; off >= 1; off >>= 1) s += __shfl_xor(s, off, 32);
  float inv = 1.0f / s;
  for (int i = 0; i < 8; ++i) p[lane + 32 * i] = v[i] * inv;
}

// ---------------- pinv scale: global max column-sum ----------------
__global__ void scale_init_kernel(float* s) {
  if (threadIdx.x == 0 && blockIdx.x == 0) *(int*)s = 0;
}
__global__ __launch_bounds__(256) void colsum_max_kernel(const float* __restrict__ a, float* s) {
  int idx = blockIdx.x * blockDim.x + threadIdx.x;      // kBH*256
  int col = idx & 255, z = idx >> 8;
  const float* p = a + (size_t)z * 65536 + col;
  float sum = 0.f;
  for (int r = 0; r < 256; ++r) sum += p[r * 256];
  atomicMax((int*)s, __float_as_int(sum));              // row-softmax => positive
}
__global__ __launch_bounds__(256) void zinit_kernel(const float* __restrict__ a,
                                                    const float* __restrict__ s,
                                                    float* __restrict__ zo) {
  int idx = blockIdx.x * blockDim.x + threadIdx.x;      // kBH*65536
  int j = idx & 255, i = (idx >> 8) & 255, zb = idx >> 16;
  zo[idx] = a[(size_t)zb * 65536 + (size_t)j * 256 + i] / (*s);
}
__global__ __launch_bounds__(256) void ident_minus_kernel(const float* __restrict__ m,
                                                          float* __restrict__ t, float c) {
  int idx = blockIdx.x * blockDim.x + threadIdx.x;      // kBH*65536
  int j = idx & 255, i = (idx >> 8) & 255;
  t[idx] = (i == j ? c : 0.0f) - m[idx];
}

// ---------------- fused flash:  Out = softmax_rows(Q'@K'^T) @ W ----------------
// Per batch z = b*8+h, pointers = base + b*BS + h*HS + row*RS.
// 8 waves/WG, each wave owns 16 rows; two-pass softmax; PV via LDS-relayout.
__global__ __launch_bounds__(256) void flash_kernel(
    const float* __restrict__ Qb, long qBS, long qHS, long qRS, float qscale,
    const float* __restrict__ Kb, long kBS, long kHS, long kRS,
    const float* __restrict__ Wb, long wBS, long wHS, long wRS,
    float* __restrict__ Ob, long oBS, long oHS, long oRS,
    int rows, int cols) {
  __shared__ float ptile[8][16][17];
  int z = blockIdx.z, b = z >> 3, h = z & 7;
  const float* Qp = Qb + b * qBS + h * qHS;
  const float* Kp = Kb + b * kBS + h * kHS;
  const float* Wp = Wb + b * wBS + h * wHS;
  float* Op = Ob + b * oBS + h * oHS;
  int wave = threadIdx.x >> 5, lane = threadIdx.x & 31;
  int l15 = lane & 15, hi2 = (lane >> 4) * 2;
  int row0 = blockIdx.x * 128 + wave * 16;
  if (row0 >= rows) return;
  // preload A fragments of Q' (K=64 -> 16 fp32 k-steps)
  v2f aq[16];
  {
    const float* qr = Qp + (long)(row0 + l15) * qRS;
    for (int ks = 0; ks < 16; ++ks) {
      int ko = ks * 4 + hi2;
      v2f t = { qr[ko] * qscale, qr[ko + 1] * qscale };
      aq[ks] = t;
    }
  }
  float rm[8], rs[8];
  v8f acc[4];
  for (int r = 0; r < 8; ++r) { rm[r] = -1e30f; rs[r] = 0.f; }
  for (int t = 0; t < 4; ++t) acc[t] = (v8f){};
  // ---- pass 1: row maxima ----
  for (int c0 = 0; c0 < cols; c0 += 16) {
    v8f s = {};
    const float* kr = Kp + (long)(c0 + l15) * kRS;
    for (int ks = 0; ks < 16; ++ks) {
      int ko = ks * 4 + hi2;
      v2f bf = { kr[ko], kr[ko + 1] };
      s = wmma_f32(aq[ks], bf, s);
    }
    for (int r = 0; r < 8; ++r) {
      float m = s[r];
      for (int off = 1; off < 16; off <<= 1) m = fmaxf(m, __shfl_xor(m, off, 32));
      rm[r] = fmaxf(rm[r], m);
    }
  }
  // ---- pass 2: p = exp(s - rm); rowsum; acc += P @ W ----
  for (int c0 = 0; c0 < cols; c0 += 16) {
    v8f s = {};
    const float* kr = Kp + (long)(c0 + l15) * kRS;
    for (int ks = 0; ks < 16; ++ks) {
      int ko = ks * 4 + hi2;
      v2f bf = { kr[ko], kr[ko + 1] };
      s = wmma_f32(aq[ks], bf, s);
    }
    for (int r = 0; r < 8; ++r) {
      float p = __expf(s[r] - rm[r]);
      float ssum = p;
      for (int off = 1; off < 16; off <<= 1) ssum += __shfl_xor(ssum, off, 32);
      rs[r] += ssum;
      ptile[wave][r + hi2 * 4][l15] = p;   // C-layout -> LDS (per-wave tile)
    }
    for (int kk = 0; kk < 4; ++kk) {
      int ks = kk * 4 + hi2;
      v2f ap = { ptile[wave][l15][ks], ptile[wave][l15][ks + 1] };  // A-layout
      const float* wr0 = Wp + (long)(c0 + ks) * wRS;
      const float* wr1 = Wp + (long)(c0 + ks + 1) * wRS;
      for (int t = 0; t < 4; ++t) {
        v2f bw = { wr0[t * 16 + l15], wr1[t * 16 + l15] };
        acc[t] = wmma_f32(ap, bw, acc[t]);
      }
    }
  }
  for (int t = 0; t < 4; ++t)
    for (int r = 0; r < 8; ++r) {
      int grow = row0 + r + hi2 * 4;
      Op[(long)grow * oRS + t * 16 + l15] = acc[t][r] / rs[r];
    }
}

// ---------------- depthwise conv residual: attn += conv33(v) ----------------
__global__ __launch_bounds__(256) void conv_add_kernel(
    const float* __restrict__ qkv, const float* __restrict__ kern,
    float* __restrict__ attn) {
  __shared__ float vs[160][64];
  __shared__ float kk[kKER];
  int z = blockIdx.z, b = z >> 3, h = z & 7;
  int n0 = blockIdx.x * 128;
  int tid = threadIdx.x;
  if (tid < kKER) kk[tid] = kern[h * kKER + tid];
  for (int i = tid; i < 160 * 64; i += 256) {
    int rr = i >> 6, d = i & 63;
    int np = n0 - 16 + rr;
    if (np >= 0 && np < kN) {
      async_load_b32(qkv + ((size_t)(b * kN + np)) * 1536 + 1024 + h * 64 + d,
                     &vs[rr][d]);
    } else {
      vs[rr][d] = 0.f;
    }
  }
  wait_async0();
  __syncthreads();
  int d = tid & 63, r0 = tid >> 6;
  for (int nl = r0; nl < 128; nl += 4) {
    float s = 0.f;
    for (int j = 0; j < kKER; ++j) s += kk[j] * vs[nl + j][d];
    size_t o = ((size_t)(b * kN + n0 + nl)) * kDIM + h * 64 + d;
    attn[o] += s;
  }
}

extern "C" void kernel_launch(void* const* d_in, const int* in_sizes, int n_in,
                              void* d_out, int out_size, void* d_ws, size_t ws_size,
                              hipStream_t stream) {
  (void)in_sizes; (void)n_in; (void)out_size; (void)ws_size;
  const float* x    = (const float*)d_in[0];
  const float* g    = (const float*)d_in[1];
  const float* bt   = (const float*)d_in[2];
  const float* wqkv = (const float*)d_in[3];
  const float* wout = (const float*)d_in[4];
  const float* bout = (const float*)d_in[5];
  const float* rker = (const float*)d_in[6];
  float* out = (float*)d_out;
  float* ws  = (float*)d_ws;

  size_t off = 0;
  float* XN   = ws + off; off += (size_t)16384 * 512;
  float* QKV  = ws + off; off += (size_t)16384 * 1536;
  float* QL   = ws + off; off += 262144;
  float* KL   = ws + off; off += 262144;
  float* KLT  = ws + off; off += 262144;
  float* A2   = ws + off; off += 1048576;
  float* Z1   = ws + off; off += 1048576;
  float* Z2   = ws + off; off += 1048576;
  float* AZ   = ws + off; off += 1048576;
  float* T1   = ws + off; off += 1048576;
  float* T2   = ws + off; off += 1048576;
  float* W3   = ws + off; off += 262144;
  float* ZW   = ws + off; off += 262144;
  float* ATTN = ws + off; off += (size_t)16384 * 512;
  float* SC   = ws + off; off += 16;

  // 1) LayerNorm
  ln_kernel<<<16384 / 8, 256, 0, stream>>>(x, g, bt, XN);

  // 2) QKV projection (16384x1536 = xn @ w_qkv)
  gemm_wmma<<<dim3(1536 / 64, 16384 / 64, 1), 256, 0, stream>>>(
      XN, wqkv, QKV, nullptr, nullptr,
      16384, 1536, 512, 512, 1536, 1536, 0, 0, 0, 0, 1.0f);

  // 3) landmark means (q_l scaled, k_l, k_l^T)
  landmark_kernel<<<(kBH * 256 * 64) / 256, 256, 0, stream>>>(QKV, QL, KL, KLT);

  // 4) sim2 = q_l @ k_l^T, then row softmax -> A2  (batch 16)
  gemm_wmma<<<dim3(256 / 64, 256 / 64, 16), 256, 0, stream>>>(
      QL, KLT, A2, nullptr, nullptr,
      256, 256, 64, 64, 256, 256, 0, 16384, 16384, 65536, 1.0f);
  softmax256_kernel<<<(16 * 256) / 8, 256, 0, stream>>>(A2);

  // 5) Newton-Schulz pinv (fp32 for numerical fidelity)
  scale_init_kernel<<<1, 32, 0, stream>>>(SC);
  colsum_max_kernel<<<(kBH * 256) / 256, 256, 0, stream>>>(A2, SC);
  zinit_kernel<<<(kBH * 65536) / 256, 256, 0, stream>>>(A2, SC, Z1);
  float* zc = Z1; float* zn = Z2;
  dim3 gp(256 / 64, 256 / 64, 16);
  for (int it = 0; it < 6; ++it) {
    gemm_wmma<<<gp, 256, 0, stream>>>(A2, zc, AZ, nullptr, nullptr,
        256, 256, 256, 256, 256, 256, 0, 65536, 65536, 65536, 1.0f);
    ident_minus_kernel<<<(kBH * 65536) / 256, 256, 0, stream>>>(AZ, T1, 7.0f);
    gemm_wmma<<<gp, 256, 0, stream>>>(AZ, T1, T2, nullptr, nullptr,
        256, 256, 256, 256, 256, 256, 0, 65536, 65536, 65536, 1.0f);
    ident_minus_kernel<<<(kBH * 65536) / 256, 256, 0, stream>>>(T2, T1, 15.0f);
    gemm_wmma<<<gp, 256, 0, stream>>>(AZ, T1, T2, nullptr, nullptr,
        256, 256, 256, 256, 256, 256, 0, 65536, 65536, 65536, 1.0f);
    ident_minus_kernel<<<(kBH * 65536) / 256, 256, 0, stream>>>(T2, T1, 13.0f);
    gemm_wmma<<<gp, 256, 0, stream>>>(zc, T1, zn, nullptr, nullptr,
        256, 256, 256, 256, 256, 256, 0, 65536, 65536, 65536, 0.25f);
    float* t = zc; zc = zn; zn = t;
  }

  // 6) w3 = softmax_rows(q_l @ k^T) @ v   (fused flash, rows=256, cols=8192)
  flash_kernel<<<dim3(256 / 128, 1, 16), 256, 0, stream>>>(
      QL, (long)8 * kM * kDH, (long)kM * kDH, (long)kDH, 1.0f,
      QKV + 512,  (long)kN * 1536, 64L, 1536L,
      QKV + 1024, (long)kN * 1536, 64L, 1536L,
      W3, (long)8 * kM * kDH, (long)kM * kDH, (long)kDH,
      256, 8192);

  // 7) zw = a2_inv @ w3   (batch 16, 256x64x256)
  gemm_wmma<<<dim3(64 / 64, 256 / 64, 16), 256, 0, stream>>>(
      zc, W3, ZW, nullptr, nullptr,
      256, 64, 256, 256, 64, 64, 0, 65536, 16384, 16384, 1.0f);

  // 8) attn = softmax_rows(q*scale @ k_l^T) @ zw  (fused flash, rows=8192, cols=256)
  flash_kernel<<<dim3(8192 / 128, 1, 16), 256, 0, stream>>>(
      QKV, (long)kN * 1536, 64L, 1536L, 0.125f,
      KL, (long)8 * kM * kDH, (long)kM * kDH, (long)kDH,
      ZW, (long)8 * kM * kDH, (long)kM * kDH, (long)kDH,
      ATTN, (long)kN * kDIM, 64L, (long)kDIM,
      8192, 256);

  // 9) attn += depthwise_conv33(v)
  conv_add_kernel<<<dim3(8192 / 128, 1, 16), 256, 0, stream>>>(QKV, rker, ATTN);

  // 10) out = x + attn @ w_out + b_out
  gemm_wmma<<<dim3(512 / 64, 16384 / 64, 1), 256, 0, stream>>>(
      ATTN, wout, out, bout, x,
      16384, 512, 512, 512, 512, 512, 512, 0, 0, 0, 1.0f);
}